// LSSViewTransform_50362786513389
// MI455X (gfx1250) — compile-verified
//
#include <hip/hip_runtime.h>

// ---------------- static problem config (matches reference) ----------------
#define IMG_H 48
#define IMG_W 160
#define HW    (IMG_H * IMG_W)        // 7680
#define NB    2
#define NC    128
#define ND    64
#define BEV   128
#define NPIX  (NB * HW)              // 15360
#define XMIN  (-51.2f)
#define YMIN  (-51.2f)
#define RES   (0.8f)                 // (51.2-(-51.2))/128

typedef __attribute__((ext_vector_type(2))) float v2f;
typedef __attribute__((ext_vector_type(8))) float v8f;

__global__ void lss_zero(float* __restrict__ out, int n) {
  int i = blockIdx.x * blockDim.x + threadIdx.x;
  if (i < n) out[i] = 0.0f;
}

// One wave32 per pixel; 8 waves per block. 1920 blocks x 8 = 15360 pixels exactly,
// so EXEC is all-ones everywhere (WMMA requirement).
__global__ __launch_bounds__(256) void lss_scatter(
    const float* __restrict__ feat,    // [B, C, H, W]
    const float* __restrict__ probs,   // [B, D, H, W]
    const float* __restrict__ depthv,  // [D]
    const float* __restrict__ Km,      // [B, 3, 3]
    const float* __restrict__ Tm,      // [B, 4, 4]
    float* __restrict__ out)           // [B, C, 128, 128]
{
  const int lane = threadIdx.x & 31;
  const int pix  = blockIdx.x * 8 + (threadIdx.x >> 5);
  const int b  = pix / HW;
  const int hw = pix - b * HW;
  const int h  = hw / IMG_W;
  const int w  = hw - h * IMG_W;

  // ---- K^-1 (analytic adjugate/det), wave-uniform ----
  const float* Kb = Km + b * 9;
  float a00 = Kb[0], a01 = Kb[1], a02 = Kb[2];
  float a10 = Kb[3], a11 = Kb[4], a12 = Kb[5];
  float a20 = Kb[6], a21 = Kb[7], a22 = Kb[8];
  float c00 =  (a11 * a22 - a12 * a21);
  float c01 = -(a10 * a22 - a12 * a20);
  float c02 =  (a10 * a21 - a11 * a20);
  float id  = 1.0f / (a00 * c00 + a01 * c01 + a02 * c02);
  float i00 = c00 * id;
  float i01 = -(a01 * a22 - a02 * a21) * id;
  float i02 =  (a01 * a12 - a02 * a11) * id;
  float i10 = c01 * id;
  float i11 =  (a00 * a22 - a02 * a20) * id;
  float i12 = -(a00 * a12 - a02 * a10) * id;
  float i20 = c02 * id;
  float i21 = -(a00 * a21 - a01 * a20) * id;
  float i22 =  (a00 * a11 - a01 * a10) * id;

  // ray = K^-1 * [w, h, 1]
  float gx = (float)w, gy = (float)h;
  float rx = i00 * gx + i01 * gy + i02;
  float ry = i10 * gx + i11 * gy + i12;
  float rz = i20 * gx + i21 * gy + i22;

  const float* Tb = Tm + b * 16;
  float r00 = Tb[0], r01 = Tb[1], r02 = Tb[2],  tx = Tb[3];
  float r10 = Tb[4], r11 = Tb[5], r12 = Tb[6],  ty = Tb[7];
  float r20 = Tb[8], r21 = Tb[9], r22 = Tb[10], tz = Tb[11];

  // ---- per-lane geometry for depths d = lane and d = lane+32 ----
  float    pmask[2];
  unsigned cell[2];
#pragma unroll
  for (int j = 0; j < 2; ++j) {
    int   d   = lane + 32 * j;
    float dep = depthv[d];
    float px = dep * rx, py = dep * ry, pz = dep * rz;
    float X = r00 * px + r01 * py + r02 * pz + tx;
    float Y = r10 * px + r11 * py + r12 * pz + ty;
    float Z = r20 * px + r21 * py + r22 * pz + tz;
    int bx = (int)((X - XMIN) / RES);   // trunc toward zero == astype(int32)
    int by = (int)((Y - YMIN) / RES);
    bool ok = (bx >= 0) & (bx < BEV) & (by >= 0) & (by < BEV) & (Z > 0.0f);
    float p = probs[((size_t)(b * ND + d) * IMG_H + h) * IMG_W + w];
    pmask[j] = ok ? p : 0.0f;                       // masked prob (matches reference)
    cell[j]  = ok ? (unsigned)(by * BEV + bx) : 0u; // invalid -> cell 0, value 0
  }

  // ---- per-lane feature channels c = lane + 32*j ----
  float fch[4];
#pragma unroll
  for (int j = 0; j < 4; ++j)
    fch[j] = feat[((size_t)(b * NC + lane + 32 * j) * IMG_H + h) * IMG_W + w];

  // Per-lane channel base within the [B*C, 16384] output (fits in u32: < 2^23).
  const unsigned chan0 = (unsigned)(b * NC + (lane & 15)) * (unsigned)(BEV * BEV);

  // ---- 64x128 outer product via V_WMMA_F32_16X16X4_F32, tiled 4(depth) x 8(chan) ----
#pragma unroll
  for (int dt = 0; dt < 4; ++dt) {
    // A(16x4): A[m,0] = prob[dt*16+m], all other K-columns zero.
    // Layout: VGPR0 lanes 0-15 hold (M=lane, K=0); lanes 16-31 hold K=2 -> must be 0.
    int   dsel = dt * 16 + (lane & 15);
    float ap = (dt < 2) ? __shfl(pmask[0], dsel, 32)
                        : __shfl(pmask[1], dsel - 32, 32);
    v2f A;
    A.x = (lane < 16) ? ap : 0.0f;
    A.y = 0.0f;

    // BEV cell per D-matrix row register: row r -> depth dt*16 + r + 8*(lane>=16)
    unsigned cellr[8];
#pragma unroll
    for (int r = 0; r < 8; ++r) {
      int d2 = dt * 16 + r + ((lane >= 16) ? 8 : 0);
      cellr[r] = (dt < 2) ? (unsigned)__shfl((int)cell[0], d2, 32)
                          : (unsigned)__shfl((int)cell[1], d2 - 32, 32);
    }

#pragma unroll
    for (int ct = 0; ct < 8; ++ct) {
      // B(4x16) fully replicated with feat[n], n = lane%16 -> layout-robust.
      float bval = __shfl(fch[ct >> 1], ((ct & 1) << 4) | (lane & 15), 32);
      v2f Bm;
      Bm.x = bval;
      Bm.y = bval;
      v8f Cm = {};
      // D[m][n] = prob[dt*16+m] * feat[ct*16+n]
      v8f Dm = __builtin_amdgcn_wmma_f32_16x16x4_f32(
          /*neg_a=*/false, A, /*neg_b=*/false, Bm,
          /*c_mod=*/(short)0, Cm, /*reuse_a=*/false, /*reuse_b=*/false);

      // Single u32 element offset -> SADDR-based global atomic (no 64-bit VALU math).
      const unsigned cbase = chan0 + (unsigned)(ct * 16) * (unsigned)(BEV * BEV);
#pragma unroll
      for (int r = 0; r < 8; ++r) {
        float v = Dm[r];
        if (v != 0.0f)                      // skip masked/zero contributions
          atomicAdd(out + (cbase + cellr[r]), v);
      }
    }
  }
}

extern "C" void kernel_launch(void* const* d_in, const int* in_sizes, int n_in,
                              void* d_out, int out_size, void* d_ws, size_t ws_size,
                              hipStream_t stream) {
  const float* feat   = (const float*)d_in[0];  // img_features [B,C,H,W]
  const float* probs  = (const float*)d_in[1];  // depth_probs  [B,D,H,W]
  const float* depthv = (const float*)d_in[2];  // depth_values [D]
  const float* Km     = (const float*)d_in[3];  // K [B,3,3]
  const float* Tm     = (const float*)d_in[4];  // T [B,4,4]
  float* out = (float*)d_out;                   // [B,C,128,128]

  lss_zero<<<(out_size + 255) / 256, 256, 0, stream>>>(out, out_size);
  lss_scatter<<<NPIX / 8, 256, 0, stream>>>(feat, probs, depthv, Km, Tm, out);
}